// STFT_10376640987960
// MI455X (gfx1250) — compile-verified
//
#include <hip/hip_runtime.h>

typedef float v2f __attribute__((ext_vector_type(2)));
typedef float v8f __attribute__((ext_vector_type(8)));
typedef unsigned int v4u __attribute__((ext_vector_type(4)));
typedef int v4i __attribute__((ext_vector_type(4)));
typedef int v8i __attribute__((ext_vector_type(8)));

#define FLEN    800
#define HOP     200
#define NCH     802            // 2*CUTOFF
#define BATCH   16
#define NS      441000
#define NFRAMES 2206
#define NHOPS   2209           // 441800 / 200
#define KTOT    3200           // 4 stacked frames * 800
#define NCOLS   208            // 200 padded to 13*16
#define KC      32             // K chunk staged in LDS
#define NCHUNKS (KTOT / KC)    // 100
#define LDA     34             // padded LDS strides (bank-conflict avoidance)
#define LDB     34
#define TINY32  1.1754943508222875e-38f

#if defined(__has_builtin)
#if __has_builtin(__builtin_amdgcn_tensor_load_to_lds) && \
    __has_builtin(__builtin_amdgcn_s_wait_tensorcnt)
#define USE_TDM 1
#endif
#endif
#ifndef USE_TDM
#define USE_TDM 0
#endif

// ---------------------------------------------------------------------------
// Kernel 1: Ghat_t[c][m] = (c<200) ? sum_ch Wf[ch][m%800] * Wi[ch][200*(m/800)+c] : 0
// Register-tiled 2(m) x 4(c): wf pair reused across 4 columns, one float4 wi
// load reused across 2 rows -> ~4x fewer memory instructions than naive.
// ---------------------------------------------------------------------------
__global__ __launch_bounds__(256)
void build_ghat(const float* __restrict__ wf, const float* __restrict__ wi,
                float* __restrict__ ghat) {
  const int MG = KTOT / 2;                 // 1600 m-pairs
  int gid = blockIdx.x * 256 + threadIdx.x;
  if (gid >= MG * (NCOLS / 4)) return;
  int cg = gid / MG;
  int mg = gid - cg * MG;
  int m0 = mg * 2, c0 = cg * 4;
  int j = m0 / 800, kp = m0 - j * 800;     // m0 even -> m0,m0+1 share j
  float s00 = 0, s01 = 0, s02 = 0, s03 = 0, s10 = 0, s11 = 0, s12 = 0, s13 = 0;
  if (c0 < 200) {                          // cols 200..207 are zero padding
    int c2 = 200 * j + c0;                 // c0%4==0 -> float4-aligned
    const float* pf = wf + kp;
    const float* pi = wi + c2;
#pragma unroll 2
    for (int ch = 0; ch < NCH; ++ch) {
      float a0 = pf[0], a1 = pf[1];
      float4 bb = *(const float4*)pi;
      s00 = fmaf(a0, bb.x, s00); s01 = fmaf(a0, bb.y, s01);
      s02 = fmaf(a0, bb.z, s02); s03 = fmaf(a0, bb.w, s03);
      s10 = fmaf(a1, bb.x, s10); s11 = fmaf(a1, bb.y, s11);
      s12 = fmaf(a1, bb.z, s12); s13 = fmaf(a1, bb.w, s13);
      pf += 800; pi += 800;
    }
  }
  ghat[(long)(c0 + 0) * KTOT + m0]     = s00;
  ghat[(long)(c0 + 1) * KTOT + m0]     = s01;
  ghat[(long)(c0 + 2) * KTOT + m0]     = s02;
  ghat[(long)(c0 + 3) * KTOT + m0]     = s03;
  ghat[(long)(c0 + 0) * KTOT + m0 + 1] = s10;
  ghat[(long)(c0 + 1) * KTOT + m0 + 1] = s11;
  ghat[(long)(c0 + 2) * KTOT + m0 + 1] = s12;
  ghat[(long)(c0 + 3) * KTOT + m0 + 1] = s13;
}

#if USE_TDM
// ---------------------------------------------------------------------------
// TDM descriptor (CDNA5 D#, CDNA5 ISA): load a 2D tile of
// KC x NCOLS f32 from Ghat_t (row stride KTOT elements) into LDS, with
// hardware LDS padding of 2 DWORDs every 32 DWORDs (-> row stride LDB=34).
// This toolchain's builtin takes 6 args: (g0, g1, g2, g3, g4, cpol); the
// trailing int32x8 is zero-filled (count=1 descriptor, single tile).
// ---------------------------------------------------------------------------
__device__ __forceinline__ void tdm_load_b(const float* gsrc, unsigned lds_off) {
  unsigned long long ga = (unsigned long long)(uintptr_t)gsrc;
  v4u g0;
  g0.x = 1u;                                       // count=1, is_restore=0, no gather
  g0.y = lds_off;                                  // lds_addr (bytes, wave-relative)
  g0.z = (unsigned)(ga & 0xFFFFFFFFu);             // global_addr[31:0]
  g0.w = (unsigned)((ga >> 32) & 0x01FFFFFFu)      // global_addr[56:32]
       | (2u << 30);                               // type=2 ("image")
  v8i g1;
  g1[0] = (int)((2u << 16)                         // data_size = 4B
              | (1u << 20)                         // pad_enable
              | (4u << 22)                         // pad_interval: 32 DWORDs
              | (1u << 25));                       // pad_amount: 2 DWORDs
  g1[1] = (int)(((unsigned)KTOT & 0xFFFFu) << 16); // tensor_dim0[15:0] @ [63:48]
  g1[2] = (int)((unsigned)NCOLS << 16);            // tensor_dim1[15:0] @ [95:80]
  g1[3] = (int)((unsigned)KC << 16);               // tile_dim0 @ [127:112]
  g1[4] = (int)NCOLS;                              // tile_dim1 @ [143:128]
  g1[5] = (int)KTOT;                               // tensor_dim0_stride[31:0]
  g1[6] = 0;
  g1[7] = 0;
  v4i g2 = {0, 0, 0, 0};                           // 2D tile: dims 2-4 unused
  v4i g3 = {0, 0, 0, 0};
  v8i g4 = {0, 0, 0, 0, 0, 0, 0, 0};               // unused trailing group
  __builtin_amdgcn_tensor_load_to_lds(g0, g1, g2, g3, g4, 0);
}
#endif

// ---------------------------------------------------------------------------
// Kernel 2: per batch, Y(2209 x 200) = U(2209 x 3200) @ Ghat(3200 x 200)
// with U[q, 800j+k'] = xpad[200(q-j)+k'] generated inline (reflect pad +
// frame validity), then window-sum normalize, x4 scale, crop.
// Software-pipelined: A prefetched into registers, B via double-buffered
// TDM (tensor_load_to_lds + s_wait_tensorcnt) when available.
// ---------------------------------------------------------------------------
__global__ __launch_bounds__(256)
void stft_fused(const float* __restrict__ x, const float* __restrict__ ghat,
                const float* __restrict__ wsum, float* __restrict__ out) {
  __shared__ __align__(16) float As[128 * LDA];
  __shared__ __align__(16) float Bs[2][NCOLS * LDB];

  const int b     = blockIdx.y;
  const int q0    = blockIdx.x * 128;
  const int tid   = threadIdx.x;
  const int lane  = tid & 31;
  const int wave  = tid >> 5;
  const int ln16  = lane & 15;
  const int lhalf = lane >> 4;          // 0: K=0,1 / M=0..7   1: K=2,3 / M=8..15

  const float* xb = x + (long)b * NS;

  v8f acc[13] = {};
  float areg[16];

  auto loadA = [&](int kc0) {           // 128 x 32 tile -> registers
#pragma unroll
    for (int e = 0; e < 16; ++e) {
      int idx = tid + e * 256;
      int ci  = idx & 31, ri = idx >> 5;
      int m   = kc0 + ci;
      int j   = m / 800;
      int kp  = m - j * 800;
      int f   = (q0 + ri) - j;          // source frame index
      float v = 0.f;
      if ((unsigned)f < (unsigned)NFRAMES) {
        int p = f * HOP + kp - 400;     // reflect padding
        if (p < 0) p = -p;
        else if (p >= NS) p = 2 * NS - 2 - p;
        v = xb[p];
      }
      areg[e] = v;
    }
  };

#if !USE_TDM
  float breg[26];
  auto loadB = [&](int kc0) {           // 208 x 32 tile -> registers
#pragma unroll
    for (int e = 0; e < 26; ++e) {
      int idx = tid + e * 256;
      int k   = idx & 31, c = idx >> 5;
      breg[e] = ghat[(long)c * KTOT + kc0 + k];
    }
  };
#endif

  // ---- prologue: start chunk 0 transfers ----
  loadA(0);
#if USE_TDM
  if (wave == 0) tdm_load_b(ghat, (unsigned)(uintptr_t)&Bs[0][0]);
#else
  loadB(0);
#endif

  int cur = 0;
  for (int ch = 0; ch < NCHUNKS; ++ch) {
    const int kc0 = ch * KC;
    // commit prefetched A to LDS
#pragma unroll
    for (int e = 0; e < 16; ++e) {
      int idx = tid + e * 256;
      int ci  = idx & 31, ri = idx >> 5;
      As[ri * LDA + ci] = areg[e];
    }
#if !USE_TDM
#pragma unroll
    for (int e = 0; e < 26; ++e) {
      int idx = tid + e * 256;
      int k   = idx & 31, c = idx >> 5;
      Bs[cur][c * LDB + k] = breg[e];
    }
#endif
    // prefetch next chunk's A while this chunk computes
    if (ch + 1 < NCHUNKS) loadA(kc0 + KC);
#if USE_TDM
    if (wave == 0) __builtin_amdgcn_s_wait_tensorcnt(0);   // Bs[cur] ready
#else
    if (ch + 1 < NCHUNKS) loadB(kc0 + KC);
#endif
    __syncthreads();
#if USE_TDM
    // DMA next B tile into the other buffer, overlapped with WMMA below
    if (ch + 1 < NCHUNKS && wave == 0)
      tdm_load_b(ghat + (kc0 + KC), (unsigned)(uintptr_t)&Bs[cur ^ 1][0]);
#endif

    // ---- WMMA: A frags reused across all 13 N-tiles ----
    v2f af[KC / 4];
#pragma unroll
    for (int ks = 0; ks < KC / 4; ++ks)
      af[ks] = *(const v2f*)&As[(wave * 16 + ln16) * LDA + ks * 4 + 2 * lhalf];

    const float* Bc = Bs[cur];
#pragma unroll
    for (int nt = 0; nt < 13; ++nt) {
#pragma unroll
      for (int ks = 0; ks < KC / 4; ++ks) {
        v2f bf = *(const v2f*)&Bc[(nt * 16 + ln16) * LDB + ks * 4 + 2 * lhalf];
        acc[nt] = __builtin_amdgcn_wmma_f32_16x16x4_f32(
            false, af[ks], false, bf, (short)0, acc[nt], false, false);
      }
    }
    __syncthreads();
#if USE_TDM
    cur ^= 1;
#endif
  }

  // ---- epilogue: window-sum normalize, x4 scale, crop, store once ----
  const int qw = q0 + wave * 16;
#pragma unroll
  for (int nt = 0; nt < 13; ++nt) {
    int c = nt * 16 + ln16;
    if (c >= 200) continue;
#pragma unroll
    for (int v = 0; v < 8; ++v) {
      int q = qw + v + 8 * lhalf;       // C/D layout: M = v + 8*(lane>=16)
      if (q >= NHOPS) continue;
      int t = q * HOP + c;              // index into 441800-long OLA buffer
      if (t < 400 || t >= NS + 400) continue;
      float val = acc[nt][v];
      float w   = wsum[t];
      if (w > TINY32) val /= w;
      val *= 4.0f;                      // FILTER_LENGTH / HOP_LENGTH
      out[(long)b * NS + (t - 400)] = val;
    }
  }
}

// ---------------------------------------------------------------------------
extern "C" void kernel_launch(void* const* d_in, const int* in_sizes, int n_in,
                              void* d_out, int out_size, void* d_ws, size_t ws_size,
                              hipStream_t stream) {
  const float* x   = (const float*)d_in[0];  // (16, 441000)
  const float* wf  = (const float*)d_in[1];  // (802, 800)
  const float* wi  = (const float*)d_in[2];  // (802, 800)
  const float* wsq = (const float*)d_in[3];  // (441800,)
  float* out  = (float*)d_out;               // (16, 1, 441000)
  float* ghat = (float*)d_ws;                // 208*3200 floats = 2.66 MB

  const int nthreads = (KTOT / 2) * (NCOLS / 4);   // 83200
  build_ghat<<<(nthreads + 255) / 256, 256, 0, stream>>>(wf, wi, ghat);

  dim3 grid((NHOPS + 127) / 128, BATCH);           // (18, 16)
  stft_fused<<<grid, 256, 0, stream>>>(x, ghat, wsq, out);
}